// OnlyConv_41351945126298
// MI455X (gfx1250) — compile-verified
//
#include <hip/hip_runtime.h>

typedef __attribute__((ext_vector_type(2))) float v2f;
typedef __attribute__((ext_vector_type(8))) float v8f;
typedef __attribute__((ext_vector_type(4))) unsigned int v4u;
typedef __attribute__((ext_vector_type(4))) int v4i;
typedef __attribute__((ext_vector_type(8))) int v8i;

#define D 256
#define ROWS_PER_BLOCK 16
#define EDGES_PER_BLOCK 8

#if defined(__has_builtin)
#if __has_builtin(__builtin_amdgcn_tensor_load_to_lds)
#define HAVE_TDM 1
#endif
#endif

// ---------------------------------------------------------------------------
// Kernel 0: zero the output accumulator (harness poisons d_out with 0xAA).
// ---------------------------------------------------------------------------
__global__ __launch_bounds__(256) void zero_f4(float4* __restrict__ p, int n4) {
    int i = blockIdx.x * blockDim.x + threadIdx.x;
    if (i < n4) p[i] = make_float4(0.f, 0.f, 0.f, 0.f);
}

// ---------------------------------------------------------------------------
// Kernel 1: h = relu(x @ W1 + b1) @ W2 + b2, fp32 WMMA 16x16x4.
// Block = 256 threads = 8 waves. Each block: 16 rows of x, all 256 cols.
// Each wave owns two 16-col output tiles (cols wave*32 .. wave*32+31).
//
// The 16x256 f32 x tile (16 KB, contiguous in global memory) is staged into
// LDS by the Tensor Data Mover: wave 0 issues tensor_load_to_lds with a 1-D
// D# (data_size=4B, tensor_dim0=tile_dim0=4096, tile_dim1=1, type=2), gates
// on s_wait_tensorcnt 0, then the workgroup barrier publishes the tile.
// Falls back to a cooperative float4 load when the builtin is unavailable.
// ---------------------------------------------------------------------------
__global__ __launch_bounds__(256) void mlp_wmma_f32(
    const float* __restrict__ x,
    const float* __restrict__ W1, const float* __restrict__ b1,
    const float* __restrict__ W2, const float* __restrict__ b2,
    float* __restrict__ h)
{
    __shared__ float xs[ROWS_PER_BLOCK * D];   // 16 KB: x tile
    __shared__ float ts[ROWS_PER_BLOCK * D];   // 16 KB: relu(x@W1+b1) tile

    const int tid   = threadIdx.x;
    const int wave  = tid >> 5;
    const int lane  = tid & 31;
    const int row0  = blockIdx.x * ROWS_PER_BLOCK;
    const int ml    = lane & 15;          // m (A) / n (B) / col (C)
    const int kh    = (lane >> 4) << 1;   // 0 for lanes 0-15, 2 for lanes 16-31
    const int rbase = (lane >> 4) << 3;   // C/D row offset: 0 or 8
    const int col0  = wave * 32;

#ifdef HAVE_TDM
    // ---- TDM stage of the x tile: one DMA op issued by wave 0 ----
    if (wave == 0) {
        const unsigned long long gaddr =
            (unsigned long long)(const void*)(x + (size_t)row0 * D);
        const unsigned int lds_off = (unsigned int)(size_t)(&xs[0]);

        v4u g0 = (v4u)0;
        g0.x = 1u;                                   // count=1, user descriptor
        g0.y = lds_off;                              // lds_addr (bytes)
        g0.z = (unsigned int)gaddr;                  // global_addr[31:0]
        g0.w = ((unsigned int)(gaddr >> 32) & 0x01FFFFFFu)
             | (2u << 30);                           // global_addr[56:32] | type=2

        v8i g1 = (v8i)0;
        g1[0] = 0x00020000;                          // data_size=2 -> 4 bytes
        g1[1] = (int)(4096u << 16);                  // tensor_dim0[15:0]=4096
        g1[2] = (int)(1u << 16);                     // tensor_dim0 hi=0, tensor_dim1=1
        g1[3] = (int)(4096u << 16);                  // tile_dim0=4096
        g1[4] = 1;                                   // tile_dim1=1, tile_dim2=0
        g1[5] = 4096;                                // tensor_dim0_stride lo
        // g1[6], g1[7] = 0                          // stride hi / dim1_stride

        v4i gz = (v4i)0;
#if defined(__clang_major__) && (__clang_major__ >= 23)
        v8i gz8 = (v8i)0;
        __builtin_amdgcn_tensor_load_to_lds(g0, g1, gz, gz, gz8, 0);
#else
        __builtin_amdgcn_tensor_load_to_lds(g0, g1, gz, gz, 0);
#endif
        __builtin_amdgcn_s_wait_tensorcnt(0);
    }
#else
    // Cooperative load of the 16x256 x tile (1024 float4, 4 per thread).
    {
        const float4* xg = (const float4*)(x + (size_t)row0 * D);
        float4*       xl = (float4*)xs;
        #pragma unroll
        for (int i = 0; i < (ROWS_PER_BLOCK * D / 4) / 256; ++i)
            xl[tid + i * 256] = xg[tid + i * 256];
    }
#endif
    __syncthreads();

    // ---- GEMM 1: t = x @ W1 ----
    v8f c0 = {}; v8f c1 = {};
    #pragma unroll 4
    for (int k = 0; k < D; k += 4) {
        v2f a;
        a.x = xs[ml * D + k + kh + 0];
        a.y = xs[ml * D + k + kh + 1];
        v2f bA, bB;
        bA.x = W1[(k + kh + 0) * D + col0 + ml];
        bA.y = W1[(k + kh + 1) * D + col0 + ml];
        bB.x = W1[(k + kh + 0) * D + col0 + 16 + ml];
        bB.y = W1[(k + kh + 1) * D + col0 + 16 + ml];
        c0 = __builtin_amdgcn_wmma_f32_16x16x4_f32(false, a, false, bA,
                                                   (short)0, c0, false, false);
        c1 = __builtin_amdgcn_wmma_f32_16x16x4_f32(false, a, false, bB,
                                                   (short)0, c1, false, false);
    }

    // bias + ReLU -> LDS
    {
        const float bb0 = b1[col0 + ml];
        const float bb1 = b1[col0 + 16 + ml];
        #pragma unroll
        for (int r = 0; r < 8; ++r) {
            float v0 = c0[r] + bb0; v0 = v0 > 0.f ? v0 : 0.f;
            float v1 = c1[r] + bb1; v1 = v1 > 0.f ? v1 : 0.f;
            ts[(rbase + r) * D + col0 + ml]      = v0;
            ts[(rbase + r) * D + col0 + 16 + ml] = v1;
        }
    }
    __syncthreads();

    // ---- GEMM 2: h = t @ W2 ----
    v8f d0 = {}; v8f d1 = {};
    #pragma unroll 4
    for (int k = 0; k < D; k += 4) {
        v2f a;
        a.x = ts[ml * D + k + kh + 0];
        a.y = ts[ml * D + k + kh + 1];
        v2f bA, bB;
        bA.x = W2[(k + kh + 0) * D + col0 + ml];
        bA.y = W2[(k + kh + 1) * D + col0 + ml];
        bB.x = W2[(k + kh + 0) * D + col0 + 16 + ml];
        bB.y = W2[(k + kh + 1) * D + col0 + 16 + ml];
        d0 = __builtin_amdgcn_wmma_f32_16x16x4_f32(false, a, false, bA,
                                                   (short)0, d0, false, false);
        d1 = __builtin_amdgcn_wmma_f32_16x16x4_f32(false, a, false, bB,
                                                   (short)0, d1, false, false);
    }

    // bias + store h tile
    {
        const float bb0 = b2[col0 + ml];
        const float bb1 = b2[col0 + 16 + ml];
        #pragma unroll
        for (int r = 0; r < 8; ++r) {
            const size_t row = (size_t)(row0 + rbase + r);
            h[row * D + col0 + ml]      = d0[r] + bb0;
            h[row * D + col0 + 16 + ml] = d1[r] + bb1;
        }
    }
}

// ---------------------------------------------------------------------------
// Kernel 2: out[row] += h[col] over the COO edge list.
// One wave32 per edge; each lane gathers 2 float4 (8 floats) of h[col] and
// scatter-adds into out[row] with native f32 global atomics. h and out both
// fit in the 192 MB L2, so gathers + atomics stay on-chip.
// ---------------------------------------------------------------------------
__global__ __launch_bounds__(256) void spmm_scatter(
    const int* __restrict__ erow, const int* __restrict__ ecol,
    const float* __restrict__ h, float* __restrict__ out, int n_edges)
{
    const int wave = threadIdx.x >> 5;
    const int lane = threadIdx.x & 31;
    const int e    = blockIdx.x * EDGES_PER_BLOCK + wave;
    if (e >= n_edges) return;

    const int col = ecol[e];
    const int row = erow[e];

    // Prefetch the next edge's source row into cache (global_prefetch_b8).
    if (e + EDGES_PER_BLOCK < n_edges) {
        const int ncol = ecol[e + EDGES_PER_BLOCK];
        __builtin_prefetch(h + (size_t)ncol * D + lane * 8, 0, 0);
    }

    const float4* src = (const float4*)(h + (size_t)col * D) + lane;
    float*        dst = out + (size_t)row * D;

    const float4 v0 = src[0];     // floats [lane*4,      lane*4+3]
    const float4 v1 = src[32];    // floats [128+lane*4,  128+lane*4+3]

    const int c = lane * 4;
    unsafeAtomicAdd(dst + c + 0,       v0.x);
    unsafeAtomicAdd(dst + c + 1,       v0.y);
    unsafeAtomicAdd(dst + c + 2,       v0.z);
    unsafeAtomicAdd(dst + c + 3,       v0.w);
    unsafeAtomicAdd(dst + c + 128 + 0, v1.x);
    unsafeAtomicAdd(dst + c + 128 + 1, v1.y);
    unsafeAtomicAdd(dst + c + 128 + 2, v1.z);
    unsafeAtomicAdd(dst + c + 128 + 3, v1.w);
}

// ---------------------------------------------------------------------------
// Launch
// ---------------------------------------------------------------------------
extern "C" void kernel_launch(void* const* d_in, const int* in_sizes, int n_in,
                              void* d_out, int out_size, void* d_ws, size_t ws_size,
                              hipStream_t stream) {
    const float* x  = (const float*)d_in[0];
    const float* W1 = (const float*)d_in[1];
    const float* b1 = (const float*)d_in[2];
    const float* W2 = (const float*)d_in[3];
    const float* b2 = (const float*)d_in[4];
    const int* erow = (const int*)d_in[5];
    const int* ecol = (const int*)d_in[6];

    const int n_nodes = in_sizes[0] / D;
    const int n_edges = in_sizes[5];

    float* h   = (float*)d_ws;          // [n_nodes, D] fp32 intermediate
    float* out = (float*)d_out;

    // 1) zero accumulator
    {
        const int n4 = (n_nodes * D) / 4;
        zero_f4<<<(n4 + 255) / 256, 256, 0, stream>>>((float4*)out, n4);
    }
    // 2) MLP: h = relu(x@W1+b1)@W2+b2   (n_nodes = 100000 is divisible by 16)
    mlp_wmma_f32<<<n_nodes / ROWS_PER_BLOCK, 256, 0, stream>>>(x, W1, b1, W2, b2, h);
    // 3) scatter-add over edges
    spmm_scatter<<<(n_edges + EDGES_PER_BLOCK - 1) / EDGES_PER_BLOCK, 256, 0, stream>>>(
        erow, ecol, h, out, n_edges);
}